// ZBLRepulsion_20650202759277
// MI455X (gfx1250) — compile-verified
//
#include <hip/hip_runtime.h>

// ---------------------------------------------------------------------------
// ZBL repulsion: per-edge gather -> transcendental pointwise -> scatter-add.
// Memory-bound: 16 B/edge streamed (~102 MB -> ~4.4 us at 23.3 TB/s HBM);
// node gathers + output accumulator are L2-resident (<= 1 MB total).
// Strategy:
//   k0: 1-thread param kernel (softplus + c-normalization) -> d_ws[0..15]
//   k1: per-node table {z, z^p}  -> d_ws+64  (kills 2 log + 2 exp per edge,
//       and fuses two b32 gathers into one b64 gather per endpoint)
//   k2: zero the output accumulator
//   k3: edge kernel: b128 streaming loads, gfx1250 global_prefetch_b8,
//       b64 node-table gathers, non-returning global_atomic_add_f32 scatter.
// ---------------------------------------------------------------------------

#define KE_F 14.399645351950548f

struct ZblParams {
    float a1, a2, a3, a4;
    float c1, c2, c3, c4;
    float p, d;
};

__device__ __forceinline__ float softplus_f(float x) {
    // numerically stable softplus: max(x,0) + log1p(exp(-|x|))
    return fmaxf(x, 0.0f) + log1pf(expf(-fabsf(x)));
}

__device__ __forceinline__ float sigma_f(float x) {
    return (x > 0.0f) ? __expf(-1.0f / fmaxf(x, 1e-12f)) : 0.0f;
}

// ---- k0: uniform parameter preprocessing (1 thread) ------------------------
__global__ void zbl_param_kernel(
    const float* __restrict__ a1r, const float* __restrict__ a2r,
    const float* __restrict__ a3r, const float* __restrict__ a4r,
    const float* __restrict__ c1r, const float* __restrict__ c2r,
    const float* __restrict__ c3r, const float* __restrict__ c4r,
    const float* __restrict__ pr,  const float* __restrict__ dr,
    float* __restrict__ wsp)
{
    if (threadIdx.x != 0 || blockIdx.x != 0) return;
    float a1 = softplus_f(a1r[0]);
    float a2 = softplus_f(a2r[0]);
    float a3 = softplus_f(a3r[0]);
    float a4 = softplus_f(a4r[0]);
    float c1 = softplus_f(c1r[0]);
    float c2 = softplus_f(c2r[0]);
    float c3 = softplus_f(c3r[0]);
    float c4 = softplus_f(c4r[0]);
    float inv = 1.0f / (c1 + c2 + c3 + c4);
    wsp[0] = a1; wsp[1] = a2; wsp[2] = a3; wsp[3] = a4;
    wsp[4] = c1 * inv; wsp[5] = c2 * inv; wsp[6] = c3 * inv; wsp[7] = c4 * inv;
    wsp[8] = softplus_f(pr[0]);
    wsp[9] = softplus_f(dr[0]);
}

// ---- k1: per-node table {z, z^p} ------------------------------------------
__global__ __launch_bounds__(256) void zbl_node_table_kernel(
    const float* __restrict__ an,
    const float* __restrict__ wsp,
    float2* __restrict__ table, int N)
{
    int i = blockIdx.x * blockDim.x + threadIdx.x;
    if (i >= N) return;
    float p = wsp[8];           // uniform -> scalar load
    float z = an[i];
    table[i] = make_float2(z, __powf(z, p));
}

// ---- k2: zero accumulator --------------------------------------------------
__global__ __launch_bounds__(256) void zbl_zero_kernel(float* __restrict__ out, int n) {
    int i = blockIdx.x * blockDim.x + threadIdx.x;
    if (i < n) out[i] = 0.0f;
}

// ---- shared per-edge math (table path: z^p precomputed) --------------------
__device__ __forceinline__ float edge_energy_tab(float zi, float zpi,
                                                 float zj, float zpj,
                                                 float cut, float L,
                                                 const ZblParams& P) {
    float x   = KE_F * cut * (zi * zj) / fmaxf(L, 1e-6f);
    float rzd = L * (zpi + zpj) * P.d;
    float y   = P.c1 * __expf(-P.a1 * rzd)
              + P.c2 * __expf(-P.a2 * rzd)
              + P.c3 * __expf(-P.a3 * rzd)
              + P.c4 * __expf(-P.a4 * rzd);
    float c  = L * (1.0f / 1.5f);          // switch(L, 0, 1.5)
    float s1 = sigma_f(1.0f - c);
    float s0 = sigma_f(c);
    float w  = s1 / (s1 + s0 + 1e-12f);
    return 0.5f * w * x * y;
}

// ---- k3: main edge kernel (table path) -------------------------------------
__global__ __launch_bounds__(256) void zbl_edge_kernel_tab(
    const float2* __restrict__ table,  // {z, z^p} per node
    const float*  __restrict__ cut,
    const int*    __restrict__ snd,
    const int*    __restrict__ rcv,
    const float*  __restrict__ len,
    const float*  __restrict__ wsp,    // 10 uniform params
    float* __restrict__ out,
    int E)
{
    ZblParams P;
    P.a1 = wsp[0]; P.a2 = wsp[1]; P.a3 = wsp[2]; P.a4 = wsp[3];
    P.c1 = wsp[4]; P.c2 = wsp[5]; P.c3 = wsp[6]; P.c4 = wsp[7];
    P.p  = wsp[8]; P.d  = wsp[9];

    int gid  = blockIdx.x * blockDim.x + threadIdx.x;
    int base = gid * 4;
    if (base >= E) return;

    // gfx1250 prefetch one block-width (256 thr * 4 elems = 16 KB) ahead.
    int pf = base + 256 * 4;
    if (pf < E) {
        __builtin_prefetch(cut + pf, 0, 1);
        __builtin_prefetch(len + pf, 0, 1);
        __builtin_prefetch(snd + pf, 0, 1);
        __builtin_prefetch(rcv + pf, 0, 1);
    }

    if (base + 3 < E) {
        float4 cu = *reinterpret_cast<const float4*>(cut + base);
        float4 ll = *reinterpret_cast<const float4*>(len + base);
        int4   ss = *reinterpret_cast<const int4*>(snd + base);
        int4   rr = *reinterpret_cast<const int4*>(rcv + base);

        // b64 gathers, L2-resident (table = 800 KB << 192 MB L2)
        float2 tj0 = table[ss.x], tj1 = table[ss.y], tj2 = table[ss.z], tj3 = table[ss.w];
        float2 ti0 = table[rr.x], ti1 = table[rr.y], ti2 = table[rr.z], ti3 = table[rr.w];

        float e0 = edge_energy_tab(ti0.x, ti0.y, tj0.x, tj0.y, cu.x, ll.x, P);
        float e1 = edge_energy_tab(ti1.x, ti1.y, tj1.x, tj1.y, cu.y, ll.y, P);
        float e2 = edge_energy_tab(ti2.x, ti2.y, tj2.x, tj2.y, cu.z, ll.z, P);
        float e3 = edge_energy_tab(ti3.x, ti3.y, tj3.x, tj3.y, cu.w, ll.w, P);

        // non-returning hardware f32 atomics into L2 (global_atomic_add_f32)
        unsafeAtomicAdd(out + rr.x, e0);
        unsafeAtomicAdd(out + rr.y, e1);
        unsafeAtomicAdd(out + rr.z, e2);
        unsafeAtomicAdd(out + rr.w, e3);
    } else {
        for (int k = 0; k < 4; ++k) {
            int e = base + k;
            if (e >= E) break;
            int s = snd[e], r = rcv[e];
            float2 tj = table[s], ti = table[r];
            float ev = edge_energy_tab(ti.x, ti.y, tj.x, tj.y, cut[e], len[e], P);
            unsafeAtomicAdd(out + r, ev);
        }
    }
}

// ---- fallback: self-contained edge kernel (no workspace needed) ------------
__global__ __launch_bounds__(256) void zbl_edge_kernel_self(
    const float* __restrict__ an,
    const float* __restrict__ cut,
    const int*   __restrict__ snd,
    const int*   __restrict__ rcv,
    const float* __restrict__ len,
    const float* __restrict__ a1r, const float* __restrict__ a2r,
    const float* __restrict__ a3r, const float* __restrict__ a4r,
    const float* __restrict__ c1r, const float* __restrict__ c2r,
    const float* __restrict__ c3r, const float* __restrict__ c4r,
    const float* __restrict__ pr,  const float* __restrict__ dr,
    float* __restrict__ out, int E)
{
    ZblParams P;
    P.a1 = softplus_f(a1r[0]); P.a2 = softplus_f(a2r[0]);
    P.a3 = softplus_f(a3r[0]); P.a4 = softplus_f(a4r[0]);
    float c1 = softplus_f(c1r[0]), c2 = softplus_f(c2r[0]);
    float c3 = softplus_f(c3r[0]), c4 = softplus_f(c4r[0]);
    float inv = 1.0f / (c1 + c2 + c3 + c4);
    P.c1 = c1 * inv; P.c2 = c2 * inv; P.c3 = c3 * inv; P.c4 = c4 * inv;
    P.p  = softplus_f(pr[0]);
    P.d  = softplus_f(dr[0]);

    int gid = blockIdx.x * blockDim.x + threadIdx.x;
    int base = gid * 4;
    for (int k = 0; k < 4; ++k) {
        int e = base + k;
        if (e >= E) break;
        int s = snd[e], r = rcv[e];
        float zi = an[r], zj = an[s];
        float ev = edge_energy_tab(zi, __powf(zi, P.p), zj, __powf(zj, P.p),
                                   cut[e], len[e], P);
        unsafeAtomicAdd(out + r, ev);
    }
}

extern "C" void kernel_launch(void* const* d_in, const int* in_sizes, int n_in,
                              void* d_out, int out_size, void* d_ws, size_t ws_size,
                              hipStream_t stream) {
    // 0: atomic_numbers [N] f32 | 1: cutoffs [E] f32 | 2: senders [E] i32
    // 3: receivers [E] i32 | 4: lengths [E] f32 | 5: num_nodes [1] i32
    // 6..15: a1..a4, c1..c4, p, d raw params [1] f32 each
    const float* an  = (const float*)d_in[0];
    const float* cut = (const float*)d_in[1];
    const int*   snd = (const int*)d_in[2];
    const int*   rcv = (const int*)d_in[3];
    const float* len = (const float*)d_in[4];
    const float* a1r = (const float*)d_in[6];
    const float* a2r = (const float*)d_in[7];
    const float* a3r = (const float*)d_in[8];
    const float* a4r = (const float*)d_in[9];
    const float* c1r = (const float*)d_in[10];
    const float* c2r = (const float*)d_in[11];
    const float* c3r = (const float*)d_in[12];
    const float* c4r = (const float*)d_in[13];
    const float* pr  = (const float*)d_in[14];
    const float* dr  = (const float*)d_in[15];
    float* out = (float*)d_out;

    int N = in_sizes[0];
    int E = in_sizes[1];

    // zero the accumulator every call (harness poisons d_out, replays graph)
    zbl_zero_kernel<<<(N + 255) / 256, 256, 0, stream>>>(out, N);

    size_t need = 64 + (size_t)N * sizeof(float2);
    if (ws_size >= need) {
        float*  wsp   = (float*)d_ws;                       // 10 params
        float2* table = (float2*)((char*)d_ws + 64);        // {z, z^p} per node
        zbl_param_kernel<<<1, 1, 0, stream>>>(
            a1r, a2r, a3r, a4r, c1r, c2r, c3r, c4r, pr, dr, wsp);
        zbl_node_table_kernel<<<(N + 255) / 256, 256, 0, stream>>>(an, wsp, table, N);
        int groups = (E + 3) / 4;
        zbl_edge_kernel_tab<<<(groups + 255) / 256, 256, 0, stream>>>(
            table, cut, snd, rcv, len, wsp, out, E);
    } else {
        int groups = (E + 3) / 4;
        zbl_edge_kernel_self<<<(groups + 255) / 256, 256, 0, stream>>>(
            an, cut, snd, rcv, len,
            a1r, a2r, a3r, a4r, c1r, c2r, c3r, c4r, pr, dr, out, E);
    }
}